// Seq_CNN_7344394076371
// MI455X (gfx1250) — compile-verified
//
#include <hip/hip_runtime.h>

// Problem constants from the reference
#define V_SZ 1000
#define O_SZ 64
#define L_SZ 64
#define B_SZ 128

typedef __attribute__((ext_vector_type(2))) float v2f;
typedef __attribute__((ext_vector_type(8))) float v8f;

// ---------------------------------------------------------------------------
// Phase 1: per-branch gather-reduce.
//   s[b,f,o] = sum_l W[o, tokens[b,l], f, l]    (W is (O,V,FH,L) row-major)
//   feat[b,o] = m/sqrt(1+m^2),  m = relu(max_f s + bias[o])
// Block = one (b, branch). 256 threads: thread -> (o = tid&63, f0 = tid>>6),
// thread accumulates f in {f0, f0+4, ...}. Token offsets staged in LDS once.
// Memory-latency bound: unroll-4 over l keeps 4..20 scattered loads in flight.
// ---------------------------------------------------------------------------
template<int FH>
__device__ __forceinline__ void branch_body(
    const int*   __restrict__ tokens,
    const float* __restrict__ W,
    const float* __restrict__ bias,
    float*       __restrict__ feats,
    int b, int branch_off,
    int* offs, float* pmax)
{
    const int tid = threadIdx.x;

    if (tid < L_SZ) {
        int t = tokens[b * L_SZ + tid];
        offs[tid] = t * (FH * L_SZ) + tid;   // v*FH*L + l
    }
    __syncthreads();

    const int o  = tid & 63;
    const int f0 = tid >> 6;                  // 0..3

    constexpr int NF = (FH + 3) / 4;
    float acc[NF];
    #pragma unroll
    for (int i = 0; i < NF; ++i) acc[i] = 0.0f;

    const float* Wo = W + (size_t)o * (V_SZ * FH * L_SZ);

    #pragma unroll 4
    for (int l = 0; l < L_SZ; ++l) {
        const int base = offs[l];
        #pragma unroll
        for (int i = 0; i < NF; ++i) {
            const int f = f0 + 4 * i;
            if (f < FH) acc[i] += Wo[base + f * L_SZ];
        }
    }

    // relu(max_f (s+bias)) == max(0, max_f acc + bias): fold ReLU through max.
    float mloc = -3.0e38f;
    #pragma unroll
    for (int i = 0; i < NF; ++i) {
        const int f = f0 + 4 * i;
        if (f < FH) mloc = fmaxf(mloc, acc[i]);
    }
    pmax[f0 * 64 + o] = mloc;
    __syncthreads();

    if (tid < 64) {
        float mm = fmaxf(fmaxf(pmax[tid], pmax[64 + tid]),
                         fmaxf(pmax[128 + tid], pmax[192 + tid]));
        float m  = fmaxf(mm + bias[tid], 0.0f);
        feats[b * 256 + branch_off + tid] = m * rsqrtf(1.0f + m * m);
    }
}

__global__ void __launch_bounds__(256)
branches_kernel(const int*   __restrict__ tokens,
                const float* __restrict__ W1, const float* __restrict__ b1,
                const float* __restrict__ W2, const float* __restrict__ b2,
                const float* __restrict__ W3, const float* __restrict__ b3,
                const float* __restrict__ W4, const float* __restrict__ b4,
                float* __restrict__ feats)
{
    __shared__ int   offs[L_SZ];
    __shared__ float pmax[256];
    const int b = blockIdx.x;
    switch (blockIdx.y) {
        case 0:  branch_body<5 >(tokens, W1, b1, feats, b,   0, offs, pmax); break;
        case 1:  branch_body<10>(tokens, W2, b2, feats, b,  64, offs, pmax); break;
        case 2:  branch_body<15>(tokens, W3, b3, feats, b, 128, offs, pmax); break;
        default: branch_body<20>(tokens, W4, b4, feats, b, 192, offs, pmax); break;
    }
}

// ---------------------------------------------------------------------------
// Phase 2: out(128x2) = feats(128x256) @ Wlin^T(256x2) + blin, via
// V_WMMA_F32_16X16X4_F32 (f32 A/B/C). One block, 8 waves; wave w owns rows
// 16w..16w+15 (M), N padded 2->16 with zero columns, K swept 256 in steps of 4.
// ISA layouts: A/B VGPR0 = K0 (lanes 0-15) / K2 (lanes 16-31), VGPR1 = K1/K3.
// D: VGPR j -> row j (lanes 0-15) / row j+8 (lanes 16-31), col = lane%16.
// EXEC stays all-ones across the WMMA loop (selects, not branches).
// ---------------------------------------------------------------------------
__global__ void __launch_bounds__(256)
linear_wmma_kernel(const float* __restrict__ feats,
                   const float* __restrict__ Wlin,
                   const float* __restrict__ blin,
                   float* __restrict__ out)
{
    __shared__ float wl[2 * 256];
    __shared__ float bl[2];
    const int tid = threadIdx.x;
    wl[tid]       = Wlin[tid];
    wl[256 + tid] = Wlin[256 + tid];
    if (tid < 2) bl[tid] = blin[tid];
    __syncthreads();

    const int lane = tid & 31;
    const int wave = tid >> 5;      // 0..7
    const int half = lane >> 4;     // 0 / 1
    const int lr   = lane & 15;
    const int row0 = wave * 16;

    v8f c = {};
    for (int kk = 0; kk < 256; kk += 4) {
        const int kbase = kk + 2 * half;

        // A: 16x4 tile of feats (M = lr, K = kbase..kbase+1)
        v2f a;
        {
            const float* p = feats + (size_t)(row0 + lr) * 256 + kbase;
            a.x = p[0];
            a.y = p[1];
        }
        // B: 4x16 tile of Wlin^T: B[k,n] = Wlin[n*256+k], 0 for n>=2.
        v2f bb;
        {
            const int n  = lr;
            const int nc = (n < 2) ? n : 0;        // clamp: load stays uniform-safe
            float vx = wl[nc * 256 + kbase];
            float vy = wl[nc * 256 + kbase + 1];
            bb.x = (n < 2) ? vx : 0.0f;            // v_cndmask, no EXEC change
            bb.y = (n < 2) ? vy : 0.0f;
        }
        c = __builtin_amdgcn_wmma_f32_16x16x4_f32(false, a, false, bb,
                                                  (short)0, c, false, false);
    }

    const int n = lr;
    if (n < 2) {
        #pragma unroll
        for (int j = 0; j < 8; ++j) {
            const int m = j + 8 * half;
            out[(size_t)(row0 + m) * 2 + n] = c[j] + bl[n];
        }
    }
}

// ---------------------------------------------------------------------------
extern "C" void kernel_launch(void* const* d_in, const int* in_sizes, int n_in,
                              void* d_out, int out_size, void* d_ws, size_t ws_size,
                              hipStream_t stream)
{
    const int*   tokens = (const int*)  d_in[0];
    const float* W1     = (const float*)d_in[1];
    const float* b1     = (const float*)d_in[2];
    const float* W2     = (const float*)d_in[3];
    const float* b2     = (const float*)d_in[4];
    const float* W3     = (const float*)d_in[5];
    const float* b3     = (const float*)d_in[6];
    const float* W4     = (const float*)d_in[7];
    const float* b4     = (const float*)d_in[8];
    const float* Wlin   = (const float*)d_in[9];
    const float* blin   = (const float*)d_in[10];
    float*       out    = (float*)d_out;
    float*       feats  = (float*)d_ws;          // 128*256 f32 = 128 KB scratch

    dim3 grid(B_SZ, 4);                           // 512 blocks -> 4096 waves
    branches_kernel<<<grid, 256, 0, stream>>>(tokens, W1, b1, W2, b2,
                                              W3, b3, W4, b4, feats);
    linear_wmma_kernel<<<1, 256, 0, stream>>>(feats, Wlin, blin, out);
}